// _ECELoss_6442450944476
// MI455X (gfx1250) — compile-verified
//
#include <hip/hip_runtime.h>
#include <hip/hip_bf16.h>

// ECE (expected calibration error) over N elements, 15 bins.
// Stage 1: grid of blocks streams (conf, pred, label) tiles into LDS via the
//          CDNA5 Tensor Data Mover (double-buffered), bins into per-thread
//          LDS slots (bank-conflict-free, atomic-free), reduces per block.
// Stage 2: one block deterministically reduces per-block partials -> ECE.

#define N_BINS    15
#define TILE      1024
#define THREADS   256
#define MAX_NB    1024
#define WS_STRIDE 48   // per-block: [csum x15 | pad | cnt x15 | pad | accsum x15 | pad]

#if defined(__has_builtin)
#if __has_builtin(__builtin_amdgcn_tensor_load_to_lds) && \
    __has_builtin(__builtin_amdgcn_s_wait_tensorcnt)
#define USE_TDM 1
#else
#define USE_TDM 0
#endif
#else
#define USE_TDM 0
#endif

typedef unsigned int       u32;
typedef unsigned long long u64;
typedef __attribute__((ext_vector_type(4))) u32 v4u;
typedef __attribute__((ext_vector_type(8))) int v8i;
typedef __attribute__((ext_vector_type(4))) int v4i;

#if USE_TDM
// 1D contiguous tile load via TDM (D# per CDNA5 ISA).
// tensor_dim0 = #valid elements remaining (OOB reads return 0 -> free tail).
// 6-arg builtin (clang-23 / therock headers): (g0 v4u, g1 v8i, g2 v4i, g3 v4i,
// g4 v8i, cpol i32); groups 2/3 unused for <=2D tiles -> zeros.
__device__ __forceinline__ void tdm_load_1d(const void* gptr, u32 lds_byte_off,
                                            u32 tensor_elems, u32 tile_elems) {
  u64 ga = (u64)gptr;
  v4u g0;
  g0.x = 1u;                                              // count=1 (valid D#)
  g0.y = lds_byte_off;                                    // lds_addr
  g0.z = (u32)ga;                                         // global_addr[31:0]
  g0.w = (u32)((ga >> 32) & 0x01FFFFFFull) | (2u << 30);  // ga[56:32] | type=2
  v8i g1;
  g1[0] = (int)(2u << 16);                                // wg_mask=0, data_size=4B
  g1[1] = (int)((tensor_elems & 0xFFFFu) << 16);          // tensor_dim0[15:0]
  g1[2] = (int)((tensor_elems >> 16) & 0xFFFFu);          // tensor_dim0[31:16], dim1=0
  g1[3] = (int)(tile_elems << 16);                        // tile_dim0
  g1[4] = 0;                                              // tile_dim1=0 (1D), tile_dim2=0
  g1[5] = (int)tensor_elems;                              // tensor_dim0_stride[31:0]
  g1[6] = 0;
  g1[7] = 0;
  v4i z4 = {0, 0, 0, 0};
  v8i z8 = {0, 0, 0, 0, 0, 0, 0, 0};
  __builtin_amdgcn_tensor_load_to_lds(g0, g1, z4, z4, z8, 0);
}
#endif

__global__ __launch_bounds__(THREADS) void ece_hist_kernel(
    const float* __restrict__ conf, const int* __restrict__ pred,
    const int* __restrict__ lab, float* __restrict__ partials,
    int n, int ntiles) {
  __shared__ float sConf[2][TILE];
  __shared__ int   sPred[2][TILE];
  __shared__ int   sLab [2][TILE];
  __shared__ int   sCnt [N_BINS * THREADS];   // packed: cnt | (acc_cnt << 16)
  __shared__ float sCsum[N_BINS * THREADS];

  const int tid = threadIdx.x;
#pragma unroll
  for (int b = 0; b < N_BINS; ++b) {
    sCnt [b * THREADS + tid] = 0;
    sCsum[b * THREADS + tid] = 0.0f;
  }

  const bool wave0 = (tid >> 5) == 0;   // wave32: wave id = tid/32
  (void)wave0;
  int buf = 0;

#if USE_TDM
  {
    int t0 = (int)blockIdx.x;
    if (wave0 && t0 < ntiles) {
      u32 rem = (u32)(n - t0 * TILE);
      tdm_load_1d(conf + (size_t)t0 * TILE, (u32)(u64)&sConf[0][0], rem, TILE);
      tdm_load_1d(pred + (size_t)t0 * TILE, (u32)(u64)&sPred[0][0], rem, TILE);
      tdm_load_1d(lab  + (size_t)t0 * TILE, (u32)(u64)&sLab [0][0], rem, TILE);
    }
  }
#endif

  for (int tile = (int)blockIdx.x; tile < ntiles; tile += (int)gridDim.x) {
#if USE_TDM
    int next = tile + (int)gridDim.x;
    if (wave0) {
      if (next < ntiles) {                       // prefetch next tile (double buffer)
        u32 rem = (u32)(n - next * TILE);
        int nb2 = buf ^ 1;
        tdm_load_1d(conf + (size_t)next * TILE, (u32)(u64)&sConf[nb2][0], rem, TILE);
        tdm_load_1d(pred + (size_t)next * TILE, (u32)(u64)&sPred[nb2][0], rem, TILE);
        tdm_load_1d(lab  + (size_t)next * TILE, (u32)(u64)&sLab [nb2][0], rem, TILE);
        __builtin_amdgcn_s_wait_tensorcnt(3);    // in-order: first 3 (current tile) done
      } else {
        __builtin_amdgcn_s_wait_tensorcnt(0);
      }
    }
#else
    for (int i = tid; i < TILE; i += THREADS) {
      long long gi = (long long)tile * TILE + i;
      bool in = gi < (long long)n;
      sConf[buf][i] = in ? conf[gi] : 0.0f;
      sPred[buf][i] = in ? pred[gi] : 0;
      sLab [buf][i] = in ? lab [gi] : 0;
    }
#endif
    __syncthreads();

#pragma unroll
    for (int k = 0; k < TILE / THREADS; ++k) {
      int i = k * THREADS + tid;
      float c = sConf[buf][i];
      // searchsorted(boundaries, c, 'left') - 1 with boundaries[k] = k/15:
      float t = c * 15.0f;
      int bin = (int)t;                 // floor (c >= 0)
      const float S = 0.066666667f;     // fp32(1/15)
      if (c <= (float)bin * S)          --bin;   // exact-boundary / rounding fixup
      else if (c > (float)(bin + 1) * S) ++bin;
      if (bin >= 0 && bin < N_BINS) {   // c <= 0 -> bin -1 -> dropped
        int inc  = (sPred[buf][i] == sLab[buf][i]) ? 0x10001 : 1;
        int slot = bin * THREADS + tid; // bank = tid%64: conflict-free
        sCnt [slot] += inc;
        sCsum[slot] += c;
      }
    }
    __syncthreads();
    buf ^= 1;
  }

  // Block reduction: 16 threads per bin, then cross-lane tree (within half-wave).
  int bin  = tid >> 4;
  int part = tid & 15;
  if (bin < N_BINS) {
    int   ci = 0;
    float cs = 0.0f;
    for (int k = part; k < THREADS; k += 16) {
      ci += sCnt [bin * THREADS + k];
      cs += sCsum[bin * THREADS + k];
    }
#pragma unroll
    for (int off = 8; off >= 1; off >>= 1) {
      ci += __shfl_xor(ci, off, 16);
      cs += __shfl_xor(cs, off, 16);
    }
    if (part == 0) {
      float* o = partials + (size_t)blockIdx.x * WS_STRIDE;
      o[bin]      = cs;                         // sum of conf
      o[16 + bin] = (float)(ci & 0xFFFF);       // count
      o[32 + bin] = (float)((u32)ci >> 16);     // count of correct
    }
  }
}

__global__ __launch_bounds__(64) void ece_finalize_kernel(
    const float* __restrict__ partials, float* __restrict__ out,
    int nblocks, float inv_n) {
  __shared__ float tot[WS_STRIDE];
  __shared__ float term[N_BINS];
  int t = threadIdx.x;
  if (t < 3 * N_BINS) {
    int g = t / N_BINS, bin = t % N_BINS;
    float s = 0.0f, comp = 0.0f;               // Kahan, fixed order -> deterministic
    for (int b = 0; b < nblocks; ++b) {
      float v  = partials[(size_t)b * WS_STRIDE + g * 16 + bin];
      float y  = v - comp;
      float tt = s + y;
      comp = (tt - s) - y;
      s = tt;
    }
    tot[g * 16 + bin] = s;
  }
  __syncthreads();
  if (t < N_BINS) {
    float cs = tot[t];
    float cn = tot[16 + t];
    float as = tot[32 + t];
    float d  = cn > 0.0f ? cn : 1.0f;
    term[t]  = cn > 0.0f ? fabsf(cs / d - as / d) * (cn * inv_n) : 0.0f;
  }
  __syncthreads();
  if (t == 0) {
    float e = 0.0f;
    for (int i = 0; i < N_BINS; ++i) e += term[i];
    out[0] = e;
  }
}

extern "C" void kernel_launch(void* const* d_in, const int* in_sizes, int n_in,
                              void* d_out, int out_size, void* d_ws, size_t ws_size,
                              hipStream_t stream) {
  (void)n_in; (void)out_size; (void)ws_size;
  const float* conf = (const float*)d_in[0];
  const int*   pred = (const int*)d_in[1];
  const int*   lab  = (const int*)d_in[2];
  int n = in_sizes[0];
  float* partials = (float*)d_ws;

  int ntiles = (n + TILE - 1) / TILE;
  if (ntiles < 1) ntiles = 1;
  int nb = ntiles < MAX_NB ? ntiles : MAX_NB;

  ece_hist_kernel<<<nb, THREADS, 0, stream>>>(conf, pred, lab, partials, n, ntiles);
  ece_finalize_kernel<<<1, 64, 0, stream>>>(partials, (float*)d_out, nb,
                                            1.0f / (float)n);
}